// GCNModel_31121333027185
// MI455X (gfx1250) — compile-verified
//
#include <hip/hip_runtime.h>
#include <math.h>

#define FDIM 128
#define BN_EPS 1e-5f

typedef float v2f __attribute__((ext_vector_type(2)));
typedef float v8f __attribute__((ext_vector_type(8)));

// ---------------- degree / norm ----------------
__global__ void k_init_deg(float* __restrict__ deg, int n) {
    int i = blockIdx.x * blockDim.x + threadIdx.x;
    if (i < n) deg[i] = 1.0f;                       // +1 self loop
}

__global__ void k_count_deg(const int* __restrict__ dst, float* __restrict__ deg, int e) {
    int i = blockIdx.x * blockDim.x + threadIdx.x;
    if (i < e) atomicAdd(&deg[dst[i]], 1.0f);
}

__global__ void k_rsqrt_deg(float* __restrict__ deg, int n) {
    int i = blockIdx.x * blockDim.x + threadIdx.x;
    if (i < n) deg[i] = 1.0f / sqrtf(deg[i]);       // dis = deg^{-1/2}
}

// ---------------- fp32 WMMA GEMM: O[N x 128] = H[N x 128] @ W[128 x 128] ----------------
// One wave per 16x16 output tile; block of 256 threads = 8 waves covers all 8 N-tiles
// of one 16-row M-tile. K accumulated with 32x v_wmma_f32_16x16x4_f32.
__global__ void k_gemm_wmma(const float* __restrict__ H, const float* __restrict__ W,
                            float* __restrict__ O, int nrows) {
    const int lane  = threadIdx.x & 31;
    const int nbase = (threadIdx.x >> 5) * 16;      // wave id -> output-feature tile
    const int mbase = blockIdx.x * 16;
    const int m  = lane & 15;
    const int kk = (lane >> 4) << 1;                // lanes 16-31 hold K+2,K+3

    int arow = mbase + m;
    if (arow > nrows - 1) arow = nrows - 1;         // clamp loads; EXEC stays all-ones
    const float* hrow = H + (size_t)arow * FDIM;
    const float* wcol = W + nbase + m;              // column n == lane&15 of this N-tile

    v8f acc = {};
#pragma unroll
    for (int k = 0; k < FDIM; k += 4) {
        v2f a, b;
        a.x = hrow[k + kk];
        a.y = hrow[k + kk + 1];
        b.x = wcol[(size_t)(k + kk) * FDIM];
        b.y = wcol[(size_t)(k + kk + 1) * FDIM];
        acc = __builtin_amdgcn_wmma_f32_16x16x4_f32(false, a, false, b,
                                                    (short)0, acc, false, false);
    }

    // D layout: VGPR v -> row M = v + 8*(lane>>4), col N = lane&15
    const int rh = (lane >> 4) << 3;
#pragma unroll
    for (int v = 0; v < 8; ++v) {
        int row = mbase + rh + v;
        if (row < nrows) O[(size_t)row * FDIM + nbase + m] = acc[v];
    }
}

// ---------------- agg init: agg = deg^{-1} * hw + bias ----------------
__global__ void k_agg_init(const float* __restrict__ HW, const float* __restrict__ dis,
                           const float* __restrict__ bias, float* __restrict__ AGG, int n) {
    int gid = blockIdx.x * blockDim.x + threadIdx.x;
    if (gid >= n * FDIM) return;
    int node = gid >> 7, f = gid & (FDIM - 1);
    float sn = dis[node];
    AGG[gid] = sn * sn * HW[gid] + bias[f];
}

// ---------------- edge scatter: agg[dst] += dis[s]*dis[d] * hw[src] ----------------
// One thread per (edge, 4-feature quad): 32 threads (one wave) per edge.
__global__ void k_edge_agg(const float* __restrict__ HW, const float* __restrict__ dis,
                           const int* __restrict__ src, const int* __restrict__ dst,
                           float* __restrict__ AGG, int nEdges) {
    int gid = blockIdx.x * blockDim.x + threadIdx.x;
    int e = gid >> 5;
    if (e >= nEdges) return;
    int q = (gid & 31) << 2;
    int s = src[e], d = dst[e];
    float en = dis[s] * dis[d];
    const float4 v = *(const float4*)(HW + (size_t)s * FDIM + q);
    float* o = AGG + (size_t)d * FDIM + q;
    atomicAdd(o + 0, en * v.x);
    atomicAdd(o + 1, en * v.y);
    atomicAdd(o + 2, en * v.z);
    atomicAdd(o + 3, en * v.w);
}

// ---------------- residual + BN(eval) + ReLU, in place on AGG ----------------
__global__ void k_post_bn_relu(float* __restrict__ AGG, const float* __restrict__ Hold,
                               const float* __restrict__ gamma, const float* __restrict__ beta,
                               const float* __restrict__ mean, const float* __restrict__ var,
                               int n, int residual) {
    int gid = blockIdx.x * blockDim.x + threadIdx.x;
    if (gid >= n * FDIM) return;
    int f = gid & (FDIM - 1);
    float v = AGG[gid];
    if (residual) v += Hold[gid];
    v = (v - mean[f]) * (1.0f / sqrtf(var[f] + BN_EPS)) * gamma[f] + beta[f];
    AGG[gid] = fmaxf(v, 0.0f);
}

// ---------------- final FC (128 -> 1) + sigmoid: one wave per node ----------------
__global__ void k_final_fc(const float* __restrict__ H, const float* __restrict__ fcw,
                           const float* __restrict__ fcb, float* __restrict__ out, int n) {
    int gid = blockIdx.x * blockDim.x + threadIdx.x;
    int node = gid >> 5;
    int lane = threadIdx.x & 31;
    if (node >= n) return;
    const float4 h = *(const float4*)(H + (size_t)node * FDIM + lane * 4);
    const float4 w = *(const float4*)(fcw + lane * 4);
    float s = h.x * w.x + h.y * w.y + h.z * w.z + h.w * w.w;
#pragma unroll
    for (int off = 16; off > 0; off >>= 1) s += __shfl_xor(s, off, 32);
    if (lane == 0) out[node] = 1.0f / (1.0f + expf(-(s + fcb[0])));
}

extern "C" void kernel_launch(void* const* d_in, const int* in_sizes, int n_in,
                              void* d_out, int out_size, void* d_ws, size_t ws_size,
                              hipStream_t stream) {
    (void)n_in; (void)out_size; (void)ws_size;
    const float* x     = (const float*)d_in[0];
    const int*   ei    = (const int*)d_in[1];
    const float* W     = (const float*)d_in[2];
    const float* b     = (const float*)d_in[3];
    const float* gamma = (const float*)d_in[4];
    const float* beta  = (const float*)d_in[5];
    const float* rmean = (const float*)d_in[6];
    const float* rvar  = (const float*)d_in[7];
    const float* fcw   = (const float*)d_in[8];
    const float* fcb   = (const float*)d_in[9];
    float* out = (float*)d_out;

    const int N = in_sizes[0] / FDIM;      // 50000
    const int E = in_sizes[1] / 2;         // 800000
    const int* srcIdx = ei;
    const int* dstIdx = ei + E;

    // workspace layout (floats): dis[Npad] | bufA[N*128] | bufB[N*128] | bufHW[N*128]
    float* ws    = (float*)d_ws;
    size_t npad  = ((size_t)N + 511) & ~(size_t)511;
    float* dis   = ws;
    float* bufA  = ws + npad;
    float* bufB  = bufA + (size_t)N * FDIM;
    float* bufHW = bufB + (size_t)N * FDIM;

    const int BLK = 256;
    const int gN  = (N + BLK - 1) / BLK;
    const int gE  = (E + BLK - 1) / BLK;
    const int gNF = (N * FDIM + BLK - 1) / BLK;
    const int gE32 = (int)(((size_t)E * 32 + BLK - 1) / BLK);
    const int gN32 = (int)(((size_t)N * 32 + BLK - 1) / BLK);
    const int mtiles = (N + 15) / 16;

    // degrees -> dis
    k_init_deg<<<gN, BLK, 0, stream>>>(dis, N);
    k_count_deg<<<gE, BLK, 0, stream>>>(dstIdx, dis, E);
    k_rsqrt_deg<<<gN, BLK, 0, stream>>>(dis, N);

    const float* hcur = x;
    float* aggBuf[3] = { bufA, bufB, bufA };
    for (int L = 0; L < 3; ++L) {
        k_gemm_wmma<<<mtiles, BLK, 0, stream>>>(hcur, W + (size_t)L * FDIM * FDIM, bufHW, N);
        float* agg = aggBuf[L];
        k_agg_init<<<gNF, BLK, 0, stream>>>(bufHW, dis, b + L * FDIM, agg, N);
        k_edge_agg<<<gE32, BLK, 0, stream>>>(bufHW, dis, srcIdx, dstIdx, agg, E);
        k_post_bn_relu<<<gNF, BLK, 0, stream>>>(agg, hcur,
                                                gamma + L * FDIM, beta + L * FDIM,
                                                rmean + L * FDIM, rvar + L * FDIM,
                                                N, L > 0 ? 1 : 0);
        hcur = agg;
    }

    k_final_fc<<<gN32, BLK, 0, stream>>>(hcur, fcw, fcb, out, N);
}